// MaskDINOHead_39822936768778
// MI455X (gfx1250) — compile-verified
//
#include <hip/hip_runtime.h>
#include <hip/hip_bf16.h>

#define NUM_CLASSES 80
#define NUM_QUERIES 300
#define TOPK 100
#define BATCH 2
#define HM 200
#define WM 200
#define OH 800
#define OW 800
#define RATIO 0.25f              // HM/OH = WM/OW
#define TILES_X (OW / 16)        // 50
#define TILES_PER_MASK (TILES_X * (OH / 16))   // 2500
#define WAVES_PER_BLOCK 8
#define NMASK (BATCH * TOPK)     // 200

// d_out layout (floats): boxes_scores[200*5] | labels[200] | masks[200*800*800] | keep[200]
#define OFF_LAB   ((size_t)NMASK * 5)                       // 1000
#define OFF_MASK  (OFF_LAB + (size_t)NMASK)                 // 1200
#define OFF_KEEP  (OFF_MASK + (size_t)NMASK * OH * OW)      // 1200 + 128M

typedef __attribute__((ext_vector_type(2))) float v2f;
typedef __attribute__((ext_vector_type(8))) float v8f;

// ---------------------------------------------------------------------------
// Kernel 1: per-batch stable top-100 over 24000 class logits (sigmoid is
// monotone, so top-k on logits == top-k on sigmoid). Lexicographic order
// (value desc, index asc) matches jax.lax.top_k tie-breaking. Also zeroes
// the per-mask accumulators used by the resize kernel.
// ---------------------------------------------------------------------------
__global__ void topk_kernel(const float* __restrict__ cls,
                            int* __restrict__ qidx, int* __restrict__ lab,
                            float* __restrict__ scr,
                            float* __restrict__ sumS, float* __restrict__ cnt) {
  const int b = blockIdx.x;
  const int tid = threadIdx.x;
  __shared__ float sv[256];
  __shared__ int   si[256];
  __shared__ float pV;
  __shared__ int   pI;

  if (tid < TOPK) { sumS[b * TOPK + tid] = 0.0f; cnt[b * TOPK + tid] = 0.0f; }
  if (tid == 0)   { pV = __builtin_inff(); pI = -1; }
  __syncthreads();

  const float* base = cls + (size_t)b * NUM_QUERIES * NUM_CLASSES;
  const int N = NUM_QUERIES * NUM_CLASSES;

  for (int k = 0; k < TOPK; ++k) {
    const float prevV = pV;
    const int   prevI = pI;
    float bv = -__builtin_inff();
    int   bi = 0x7fffffff;
    for (int t = tid; t < N; t += 256) {
      const float x = base[t];
      const bool adm = (x < prevV) || (x == prevV && t > prevI);
      if (adm && (x > bv || (x == bv && t < bi))) { bv = x; bi = t; }
    }
    sv[tid] = bv; si[tid] = bi;
    __syncthreads();
    for (int s = 128; s > 0; s >>= 1) {
      if (tid < s) {
        const float ov = sv[tid + s]; const int oi = si[tid + s];
        if (ov > sv[tid] || (ov == sv[tid] && oi < si[tid])) { sv[tid] = ov; si[tid] = oi; }
      }
      __syncthreads();
    }
    if (tid == 0) {
      pV = sv[0]; pI = si[0];
      const int gi = si[0];
      qidx[b * TOPK + k] = gi / NUM_CLASSES;
      lab [b * TOPK + k] = gi % NUM_CLASSES;
      scr [b * TOPK + k] = 1.0f / (1.0f + __expf(-sv[0]));
    }
    __syncthreads();
  }
}

// ---------------------------------------------------------------------------
// Kernel 2: one wave per 16x16 output tile. Vertical lerp folded into the
// WMMA A operand (16x8 patch, cols 6..7 padded); horizontal interpolation is
// a 16x16x4-f32 WMMA GEMM against the weight matrix B (rows 6..7 zero).
// D layout: VGPR g -> rows g (lanes 0-15) / g+8 (lanes 16-31), N = lane&15.
// Mask stream is written with NT stores (write-once, never re-read) so the
// 16x-reused 200x200 input rows stay resident in L2.
// ---------------------------------------------------------------------------
__global__ void __launch_bounds__(32 * WAVES_PER_BLOCK)
resize_mask_kernel(const float* __restrict__ masks,
                   const int* __restrict__ qidx,
                   float* __restrict__ out,
                   float* __restrict__ sumS, float* __restrict__ cnt) {
  const int m    = blockIdx.y;             // 0..199  (b*100 + k)
  const int b    = m / TOPK;
  const int wave = threadIdx.x >> 5;
  const int lane = threadIdx.x & 31;
  const int tile = blockIdx.x * WAVES_PER_BLOCK + wave;

  __shared__ float sAccS, sAccC;
  if (threadIdx.x == 0) { sAccS = 0.0f; sAccC = 0.0f; }
  __syncthreads();

  float myS = 0.0f, myC = 0.0f;

  if (tile < TILES_PER_MASK) {            // wave-uniform: EXEC all-1s for WMMA
    const int ty = tile / TILES_X;
    const int tx = tile % TILES_X;
    const int q  = qidx[m];
    const float* __restrict__ M = masks + ((size_t)b * NUM_QUERIES + q) * (HM * WM);

    const int r    = lane & 15;           // A row / B out-col / D col
    const int half = lane >> 4;

    // vertical interpolation for output row (ty*16 + r)
    const float sy  = ((float)(ty * 16 + r) + 0.5f) * RATIO - 0.5f;
    const float syf = floorf(sy);
    const float fy  = sy - syf;
    const int iy0c = min(max((int)syf, 0), HM - 1);
    const int iy1c = min(max((int)syf + 1, 0), HM - 1);

    // base input column for this tile: all taps land in [xb, xb+5]
    int xb = (int)floorf(((float)(tx * 16) + 0.5f) * RATIO - 0.5f);
    xb = min(max(xb, 0), WM - 6);

    // horizontal taps for output column (tx*16 + r)
    const float sx  = ((float)(tx * 16 + r) + 0.5f) * RATIO - 0.5f;
    const float sxf = floorf(sx);
    const float fx  = sx - sxf;
    const int ix0c = min(max((int)sxf, 0), WM - 1);
    const int ix1c = min(max((int)sxf + 1, 0), WM - 1);

    const float* row0 = M + (size_t)iy0c * WM;
    const float* row1 = M + (size_t)iy1c * WM;

    // A[r][j] for j in {2h, 2h+1, 2h+4, 2h+5}  (vertically blended patch)
    const int j0 = 2 * half;
    v2f A01, A23, B01, B23;
    {
      int c;
      c = min(xb + j0,     WM - 1); A01.x = (1.0f - fy) * row0[c] + fy * row1[c];
      c = min(xb + j0 + 1, WM - 1); A01.y = (1.0f - fy) * row0[c] + fy * row1[c];
      c = min(xb + j0 + 4, WM - 1); A23.x = (1.0f - fy) * row0[c] + fy * row1[c];
      c = min(xb + j0 + 5, WM - 1); A23.y = (1.0f - fy) * row0[c] + fy * row1[c];
    }
    // B[k][n]: weight of input col (xb+k) for output col n=r
    {
      int k;
      k = xb + j0;     B01.x = (1.0f - fx) * (float)(k == ix0c) + fx * (float)(k == ix1c);
      k = xb + j0 + 1; B01.y = (1.0f - fx) * (float)(k == ix0c) + fx * (float)(k == ix1c);
      k = xb + j0 + 4; B23.x = (1.0f - fx) * (float)(k == ix0c) + fx * (float)(k == ix1c);
      k = xb + j0 + 5; B23.y = (1.0f - fx) * (float)(k == ix0c) + fx * (float)(k == ix1c);
    }

    v8f acc = {0.f, 0.f, 0.f, 0.f, 0.f, 0.f, 0.f, 0.f};
    acc = __builtin_amdgcn_wmma_f32_16x16x4_f32(false, A01, false, B01,
                                                (short)0, acc, false, false);
    acc = __builtin_amdgcn_wmma_f32_16x16x4_f32(false, A23, false, B23,
                                                (short)0, acc, false, false);

    float* tileBase = out + OFF_MASK + (size_t)m * (OH * OW)
                          + (size_t)(ty * 16 + 8 * half) * OW + (tx * 16 + r);
#pragma unroll
    for (int g = 0; g < 8; ++g) {
      const float pm  = acc[g];
      const float bin = (pm > 0.0f) ? 1.0f : 0.0f;
      // NT store: mask stream is write-once, keep L2 for input reuse
      __builtin_nontemporal_store(bin, tileBase + (size_t)g * OW);
      // branchless sigmoid accumulation: v_exp + v_rcp, no EXEC games
      const float sig = __builtin_amdgcn_rcpf(1.0f + __expf(-pm));
      myS += bin * sig;
      myC += bin;
    }
  }

  // wave32 shuffle reduce, then LDS combine, then one global atomic per block
#pragma unroll
  for (int off = 16; off > 0; off >>= 1) {
    myS += __shfl_xor(myS, off, 32);
    myC += __shfl_xor(myC, off, 32);
  }
  if (lane == 0 && tile < TILES_PER_MASK) {
    atomicAdd(&sAccS, myS);
    atomicAdd(&sAccC, myC);
  }
  __syncthreads();
  if (threadIdx.x == 0) {
    atomicAdd(&sumS[m], sAccS);
    atomicAdd(&cnt[m],  sAccC);
  }
}

// ---------------------------------------------------------------------------
// Kernel 3: boxes/scores/labels/keep (200 elements)
// ---------------------------------------------------------------------------
__global__ void finalize_kernel(const float* __restrict__ boxes,
                                const int* __restrict__ qidx,
                                const int* __restrict__ lab,
                                const float* __restrict__ scr,
                                const float* __restrict__ sumS,
                                const float* __restrict__ cnt,
                                const int* __restrict__ hptr,
                                const int* __restrict__ wptr,
                                float* __restrict__ out) {
  const int i = blockIdx.x * blockDim.x + threadIdx.x;
  if (i >= NMASK) return;
  const int b = i / TOPK;
  const int q = qidx[i];
  const float fh = (float)hptr[0];
  const float fw = (float)wptr[0];
  const float* bx = boxes + ((size_t)b * NUM_QUERIES + q) * 4;
  const float cx = bx[0], cy = bx[1], bw = bx[2], bh = bx[3];

  const float c  = cnt[i];
  const float obj = sumS[i] / (c + 1e-6f);
  const float ps  = scr[i] * obj;

  const float x1 = fminf(fmaxf((cx - bw * 0.5f) * fw, 0.0f), fw - 1.0f);
  const float y1 = fminf(fmaxf((cy - bh * 0.5f) * fh, 0.0f), fh - 1.0f);
  const float x2 = fminf(fmaxf((cx + bw * 0.5f) * fw, 0.0f), fw - 1.0f);
  const float y2 = fminf(fmaxf((cy + bh * 0.5f) * fh, 0.0f), fh - 1.0f);
  const float area = (x2 - x1) * (y2 - y1);
  const bool keep = (c > 5.0f) && (area > 10.0f) && (ps > 0.05f);

  out[(size_t)i * 5 + 0] = x1;
  out[(size_t)i * 5 + 1] = y1;
  out[(size_t)i * 5 + 2] = x2;
  out[(size_t)i * 5 + 3] = y2;
  out[(size_t)i * 5 + 4] = ps;
  out[OFF_LAB  + i] = (float)lab[i];
  out[OFF_KEEP + i] = keep ? 1.0f : 0.0f;
}

extern "C" void kernel_launch(void* const* d_in, const int* in_sizes, int n_in,
                              void* d_out, int out_size, void* d_ws, size_t ws_size,
                              hipStream_t stream) {
  const float* msk   = (const float*)d_in[0];  // (2,300,200,200)
  const float* cls   = (const float*)d_in[1];  // (2,300,80)
  const float* boxes = (const float*)d_in[2];  // (2,300,4)
  const int*   h     = (const int*)d_in[3];
  const int*   w     = (const int*)d_in[4];
  float* out = (float*)d_out;

  // ~4 KB of scratch
  int*   qidx = (int*)d_ws;
  int*   lab  = qidx + NMASK;
  float* scr  = (float*)(lab + NMASK);
  float* sumS = scr + NMASK;
  float* cnt  = sumS + NMASK;

  topk_kernel<<<BATCH, 256, 0, stream>>>(cls, qidx, lab, scr, sumS, cnt);

  dim3 grid((TILES_PER_MASK + WAVES_PER_BLOCK - 1) / WAVES_PER_BLOCK, NMASK);
  resize_mask_kernel<<<grid, 32 * WAVES_PER_BLOCK, 0, stream>>>(msk, qidx, out, sumS, cnt);

  finalize_kernel<<<1, 256, 0, stream>>>(boxes, qidx, lab, scr, sumS, cnt, h, w, out);
}